// ForwardPhysicalModel_35785667510967
// MI455X (gfx1250) — compile-verified
//
#include <hip/hip_runtime.h>
#include <math.h>

// MI455X / gfx1250 Radon transform.
// - Gather-bound: 4 MB image stays resident in 192 MB L2; each lane
//   ray-marches one output column with fmaf-exact coordinates.
// - Branchless inner loop: 4 unconditional clamped gathers per sample;
//   out-of-image zeroing (cval=0) is folded into the bilinear WEIGHTS via
//   v_cndmask, so no exec-mask branches appear in the hot loop.
// - wave32 layout: lane n and lane n+16 own the same X column, disjoint Y
//   halves; partials folded with V_WMMA_F32_16X16X4_F32 (A = ones), which is
//   layout-robust: D[m][n] = sum_k B[k][n] regardless of K ordering.
// - 8 waves/block cover 8 Y-chunks; LDS folds the per-wave column sums.

typedef float v2f __attribute__((ext_vector_type(2)));
typedef float v8f __attribute__((ext_vector_type(8)));

#define N_IMG   1024
#define N_VIEWS 90
#define CENTER  512.0f   // N_IMG // 2

__global__ __launch_bounds__(256)
void radon_gfx1250_kernel(const float* __restrict__ img,
                          const float* __restrict__ theta,
                          float* __restrict__ out) {
  const int xblock = blockIdx.x;   // 0..63  : 16 output columns each
  const int angle  = blockIdx.y;   // 0..89
  const int tid    = threadIdx.x;  // 0..255
  const int wave   = tid >> 5;     // 0..7   : Y chunk of 128
  const int lane   = tid & 31;
  const int xloc   = lane & 15;    // column within block
  const int yhalf  = lane >> 4;    // 0/1    : 64-row half of the chunk

  // Per-angle rotation (uniform across block).
  const float ang = theta[angle] * 0.017453292519943295f; // deg2rad
  const float ca = cosf(ang);
  const float sa = sinf(ang);
  // skimage-radon inverse map constants:
  //   x_in = ca*X + sa*Y - c*(ca+sa-1);  y_in = -sa*X + ca*Y - c*(ca-sa-1)
  const float cx = -CENTER * (ca + sa - 1.0f);
  const float cy = -CENTER * (ca - sa - 1.0f);

  const int   X  = xblock * 16 + xloc;
  const float fX = (float)X;
  const float bx = fmaf(ca, fX, cx);   // x_in at Y=0 for this column
  const float by = fmaf(-sa, fX, cy);  // y_in at Y=0

  const int ybase = wave * 128 + yhalf * 64;

  float acc = 0.0f;
#pragma unroll 4
  for (int i = 0; i < 64; ++i) {
    const float fY  = (float)(ybase + i);
    // Recompute with fmaf every step (no accumulated drift along the ray).
    const float xin = fmaf(sa, fY, bx);
    const float yin = fmaf(ca, fY, by);

    const float x0f = floorf(xin);
    const float y0f = floorf(yin);
    const float wx  = xin - x0f;
    const float wy  = yin - y0f;
    const int x0 = (int)x0f, x1 = x0 + 1;
    const int y0 = (int)y0f, y1 = y0 + 1;

    // Clamped (always in-bounds) addresses -> loads can issue unconditionally.
    const int xc0 = min(max(x0, 0), N_IMG - 1);
    const int xc1 = min(max(x1, 0), N_IMG - 1);
    const int r0  = min(max(y0, 0), N_IMG - 1) << 10;  // row byte-offset /4
    const int r1  = min(max(y1, 0), N_IMG - 1) << 10;

    const float v00 = img[r0 + xc0];
    const float v01 = img[r0 + xc1];
    const float v10 = img[r1 + xc0];
    const float v11 = img[r1 + xc1];

    // Fold cval=0 boundary handling into the weights (v_cndmask, branchless).
    const float omwx = 1.0f - wx;
    const float omwy = 1.0f - wy;
    const float wxa = (x0 >= 0)          ? omwx : 0.0f;  // x1>=1>0 always
    const float wxb = (x1 <= N_IMG - 1)  ? wx   : 0.0f;  // x0<=1022 then
    const float wya = (y0 >= 0)          ? omwy : 0.0f;
    const float wyb = (y1 <= N_IMG - 1)  ? wy   : 0.0f;
    // Full per-tap validity: tap (y,x) needs both axes in range. Since
    // either tap may fall off either side of the image, the OOB cases per axis are
    // (x0 < 0 or x0 > N-1) hitting taps *0 and (x1 < 0 or x1 > N-1) hitting taps *1; same for y.
    // Masked 1D weights multiply out to exactly the reference's per-tap mask.
    const float wxa2 = (x0 <= N_IMG - 1) ? wxa : 0.0f;  // x0 can be > N-1 too
    const float wxb2 = (x1 >= 0)         ? wxb : 0.0f;  // x1 can be < 0 too
    const float wya2 = (y0 <= N_IMG - 1) ? wya : 0.0f;
    const float wyb2 = (y1 >= 0)         ? wyb : 0.0f;

    acc = fmaf(v00, wya2 * wxa2,
          fmaf(v01, wya2 * wxb2,
          fmaf(v10, wyb2 * wxa2,
          fmaf(v11, wyb2 * wxb2, acc))));
  }

  // --- WMMA half-fold: column n of D = acc(lane n) + acc(lane n+16) ---
  // A (16x4) = all ones, B column n = partials of lanes {n, n+16}, C = 0.
  // With A == 1, D[m][n] = sum_k B[k][n]: immune to K ordering within column.
  v2f a; a[0] = 1.0f; a[1] = 1.0f;
  v2f b; b[0] = acc;  b[1] = 0.0f;
  v8f c = {};
  // EXEC is all-ones here (no divergence before this point in the block).
  v8f d = __builtin_amdgcn_wmma_f32_16x16x4_f32(
      /*neg_a=*/false, a, /*neg_b=*/false, b,
      /*c_mod=*/(short)0, c, /*reuse_a=*/false, /*reuse_b=*/false);
  // Every lane's d[0] holds colsum[lane % 16].

  __shared__ float red[8][16];
  if (lane < 16) red[wave][xloc] = d[0];
  __syncthreads();

  if (tid < 16) {
    float s = red[0][tid];
#pragma unroll
    for (int w = 1; w < 8; ++w) s += red[w][tid];
    // Output layout: sino[X][angle], X-major (reference returns sino.T).
    out[(xblock * 16 + tid) * N_VIEWS + angle] = s;
  }
}

extern "C" void kernel_launch(void* const* d_in, const int* in_sizes, int n_in,
                              void* d_out, int out_size, void* d_ws, size_t ws_size,
                              hipStream_t stream) {
  const float* img   = (const float*)d_in[0];  // 1024*1024 f32
  const float* theta = (const float*)d_in[1];  // 90 f32 (degrees)
  float* out = (float*)d_out;                  // 1024*90 f32

  dim3 grid(N_IMG / 16, N_VIEWS);  // 64 x-blocks x 90 angles = 5760 blocks
  radon_gfx1250_kernel<<<grid, 256, 0, stream>>>(img, theta, out);
}